// QuanvLayer_54589034332920
// MI455X (gfx1250) — compile-verified
//
#include <hip/hip_runtime.h>
#include <math.h>

// ---------------------------------------------------------------------------
// QuanvLayer, analytically reduced:
//   z_k = cos(chrom[k]) * cos(pixel_k)  per 4-pixel patch
//   out0 = z1 z2 z3 ; out1 = z0 z1 ; out2 = z0 z1 z2 ; out3 = z0 z1 z2 z3
//   result[b,n,i,j] = sqrt(h[b,n,i,j]^2 + v[b,n,i,j]^2)
//     h patch: x[b, i, 4j..4j+3]   (row-major, coalesced float4)
//     v patch: x[b, 4j..4j+3, i]   (column access -> staged via async->LDS)
// Memory-bound (~48 MB traffic); no matmul structure -> no WMMA by design.
// CDNA5 path used: global_load_async_to_lds_b32 + s_wait_asynccnt.
// ---------------------------------------------------------------------------

#define HW      256          // image H == W
#define TI      64           // i-tile per block
#define RS      260          // LDS row stride (floats): 256 + 4 pad, 16B aligned
#define PLANE   (HW * TI)    // 256*64 = one (b,n) output plane

#if defined(__has_builtin)
#  if __has_builtin(__builtin_amdgcn_global_load_async_to_lds_b32) && \
      __has_builtin(__builtin_amdgcn_s_wait_asynccnt)
#    define USE_ASYNC_LDS 1
#  endif
#endif

#if USE_ASYNC_LDS
typedef __attribute__((address_space(1))) int g1_int;   // global (AS1) word
typedef __attribute__((address_space(3))) int l3_int;   // LDS (AS3) word
#endif

__global__ __launch_bounds__(256)
void quanv_kernel(const float* __restrict__ x,
                  const float* __restrict__ chrom,
                  float* __restrict__ out)
{
    __shared__ __align__(16) float vtile[TI * RS];   // vtile[c*RS + r] = x[b, r, i0+c]

    const int tid = threadIdx.x;
    const int bi  = blockIdx.x;
    const int b   = bi >> 2;
    const int i0  = (bi & 3) << 6;

    const int lane_c = tid & 63;   // fill: column c ; compute: j
    const int grp    = tid >> 6;   // 0..3

    // ---- Stage the 256x64 column tile into LDS (transpose via per-lane addr)
    const float* gcol = x + ((size_t)b << 16) + (size_t)(i0 + lane_c);
#if USE_ASYNC_LDS
    for (int it = 0; it < 64; ++it) {
        const int r = grp + (it << 2);
        __builtin_amdgcn_global_load_async_to_lds_b32(
            (g1_int*)(gcol + ((size_t)r << 8)),
            (l3_int*)&vtile[lane_c * RS + r],
            /*offset=*/0, /*cpol=*/0);
    }
    __builtin_amdgcn_s_wait_asynccnt(0);
#else
    for (int it = 0; it < 64; ++it) {
        const int r = grp + (it << 2);
        vtile[lane_c * RS + r] = gcol[(size_t)r << 8];
    }
#endif
    __syncthreads();

    // ---- Per-thread chromosome factors (uniform, cached)
    const float cc0 = __cosf(chrom[0]);
    const float cc1 = __cosf(chrom[1]);
    const float cc2 = __cosf(chrom[2]);
    const float cc3 = __cosf(chrom[3]);

    const int   j        = lane_c;
    const float* hrow0   = x   + ((size_t)b << 16) + ((size_t)i0 << 8) + (size_t)(j << 2);
    float*       outbase = out + (((size_t)b << 2) << 14) + ((size_t)i0 << 6) + (size_t)j;

    for (int t = 0; t < 16; ++t) {
        const int c = grp + (t << 2);          // 0..63, strided across the 4 groups

        // horizontal patch: row (i0+c), cols 4j..4j+3 — coalesced float4
        const float4 hp = *reinterpret_cast<const float4*>(hrow0 + ((size_t)c << 8));
        const float h0 = cc0 * __cosf(hp.x);
        const float h1 = cc1 * __cosf(hp.y);
        const float h2 = cc2 * __cosf(hp.z);
        const float h3 = cc3 * __cosf(hp.w);

        // vertical patch: col (i0+c), rows 4j..4j+3 — aligned b128 LDS read
        const float4 vp = *reinterpret_cast<const float4*>(&vtile[c * RS + (j << 2)]);
        const float v0 = cc0 * __cosf(vp.x);
        const float v1 = cc1 * __cosf(vp.y);
        const float v2 = cc2 * __cosf(vp.z);
        const float v3 = cc3 * __cosf(vp.w);

        const float ho1 = h0 * h1, ho2 = ho1 * h2, ho3 = ho2 * h3, ho0 = h1 * h2 * h3;
        const float vo1 = v0 * v1, vo2 = vo1 * v2, vo3 = vo2 * v3, vo0 = v1 * v2 * v3;

        float* o = outbase + ((size_t)c << 6);
        o[0]         = sqrtf(ho0 * ho0 + vo0 * vo0);
        o[PLANE]     = sqrtf(ho1 * ho1 + vo1 * vo1);
        o[2 * PLANE] = sqrtf(ho2 * ho2 + vo2 * vo2);
        o[3 * PLANE] = sqrtf(ho3 * ho3 + vo3 * vo3);
    }
}

extern "C" void kernel_launch(void* const* d_in, const int* in_sizes, int n_in,
                              void* d_out, int out_size, void* d_ws, size_t ws_size,
                              hipStream_t stream) {
    (void)n_in; (void)out_size; (void)d_ws; (void)ws_size;
    const float* x     = (const float*)d_in[0];   // (64,1,256,256) f32
    const float* chrom = (const float*)d_in[1];   // (4,) f32
    float*       out   = (float*)d_out;           // (64,4,256,64) f32

    const int B = in_sizes[0] >> 16;              // elements / (256*256)
    dim3 grid(B * 4), block(256);
    quanv_kernel<<<grid, block, 0, stream>>>(x, chrom, out);
}